// LIFLayerEligibility_55894704390527
// MI455X (gfx1250) — compile-verified
//
#include <hip/hip_runtime.h>
#include <hip/hip_bf16.h>
#include <math.h>

// dims (match reference)
#define Tt   100
#define Bb   128
#define INF  512
#define OUTF 512
#define BO   (Bb * OUTF)           // 65536

// LIF constants (fp32 rounding of exp(-dt/tau))
#define ALPHA 0.8187307530779818f  // exp(-1/5)  synaptic decay
#define BETA  0.9512294245007140f  // exp(-1/20) membrane decay
#define GAMMA 0.9512294245007140f  // exp(-1/20) trace decay

typedef float v2f __attribute__((ext_vector_type(2)));
typedef float v8f __attribute__((ext_vector_type(8)));

// ---------------------------------------------------------------------------
// Phase 1: C[t,b,o] = sum_i x[t,b,i] * W[o,i]   as one (T*B, IN)x(IN, OUT) GEMM
// One wave per 16x16 output tile, K swept 4 at a time with v_wmma_f32_16x16x4_f32.
// A lane layout: row = lane&15, K = vgpr + 2*(lane>>4)  -> contiguous v2f loads.
// ---------------------------------------------------------------------------
__global__ __launch_bounds__(128) void lif_gemm_xw(const float* __restrict__ X,
                                                   const float* __restrict__ W,
                                                   float* __restrict__ C) {
    const int lane = threadIdx.x & 31;
    const int wid  = blockIdx.x * 4 + (threadIdx.x >> 5);
    const int tm   = wid >> 5;      // 0..799  : M tile over T*B rows
    const int tn   = wid & 31;      // 0..31   : N tile over OUT cols
    const int half = lane >> 4;
    const int row  = lane & 15;

    const float* arow = X + (size_t)(tm * 16 + row) * INF; // spikes: row m, all K
    const float* brow = W + (size_t)(tn * 16 + row) * INF; // W[o,:] : col n, all K

    v8f acc = {};
#pragma unroll 8
    for (int k0 = 0; k0 < INF; k0 += 4) {
        __builtin_prefetch(arow + k0 + 64, 0, 1);          // global_prefetch_b8
        v2f a = *(const v2f*)(arow + k0 + 2 * half);
        v2f b = *(const v2f*)(brow + k0 + 2 * half);
        acc = __builtin_amdgcn_wmma_f32_16x16x4_f32(
            /*neg_a=*/false, a, /*neg_b=*/false, b,
            /*c_mod=*/(short)0, acc, /*reuse_a=*/false, /*reuse_b=*/false);
    }

    float* crow = C + (size_t)(tm * 16 + 8 * half) * OUTF + tn * 16 + row;
#pragma unroll
    for (int r = 0; r < 8; ++r)
        crow[(size_t)r * OUTF] = acc[r];
}

// ---------------------------------------------------------------------------
// Phase 2: per-(b,o) temporal scan. Reads C[t,b,o], emits S, U_hist (pre/post),
// and overwrites C in place with the gamma-weighted surrogate gradient
//   spw[t,b,o] = gamma^(T-1-t) / (1 + 0.03*|U_post - 1|)^2
// so Phase 3 needs no extra workspace.
// ---------------------------------------------------------------------------
__global__ __launch_bounds__(256) void lif_scan(float* __restrict__ C,
                                                float* __restrict__ S_out,
                                                float* __restrict__ U_hist) {
    const int idx = blockIdx.x * 256 + threadIdx.x;   // flat b*OUT + o, coalesced
    float I = 0.0f, U = 0.0f;
    for (int t = 0; t < Tt; ++t) {
        const size_t off = (size_t)t * BO + idx;
        float c = C[off];
        I = ALPHA * I + c;
        U = BETA * U + I;
        float S = (U >= 1.0f) ? 1.0f : 0.0f;
        S_out[off] = S;
        U_hist[(size_t)(2 * t) * BO + idx] = U;       // pre-reset snapshot
        U -= S;                                       // reset
        float du = fabsf(U - 1.0f);
        float d  = fmaf(0.03f, du, 1.0f);
        float sp = 1.0f / (d * d);                    // SuperSpike surrogate
        float wg = __expf(-0.05f * (float)(Tt - 1 - t)); // gamma^(T-1-t)
        C[off] = sp * wg;                             // in-place: becomes spw
        U_hist[(size_t)(2 * t + 1) * BO + idx] = U;   // post-reset snapshot
        U -= S;                                       // second subtraction (as in source)
    }
}

// ---------------------------------------------------------------------------
// Phase 3: trace[b,o,i] = sum_t spw[t,b,o] * x[t,b,i]
// Per-batch (512 x 100) x (100 x 512) GEMM, K = 100 = 25 wmma steps of 4.
// One wave per (b, 16x16) tile; trace is written exactly once (134 MB total).
// ---------------------------------------------------------------------------
__global__ __launch_bounds__(256) void lif_trace(const float* __restrict__ SPW,
                                                 const float* __restrict__ X,
                                                 float* __restrict__ TR) {
    const int lane = threadIdx.x & 31;
    const int wid  = blockIdx.x * 8 + (threadIdx.x >> 5);
    const int b    = wid >> 10;          // / (32*32)
    const int rem  = wid & 1023;
    const int tm   = rem >> 5;           // OUT tile
    const int tn   = rem & 31;           // IN  tile
    const int half = lane >> 4;
    const int row  = lane & 15;

    // spw laid out (t,b,o); x laid out (t,b,i): per-k stride is B*OUT / B*IN.
    const float* aptr = SPW + (size_t)b * OUTF + tm * 16 + row;
    const float* bptr = X   + (size_t)b * INF  + tn * 16 + row;
    const size_t aK = (size_t)Bb * OUTF;
    const size_t bK = (size_t)Bb * INF;

    v8f acc = {};
#pragma unroll 5
    for (int k0 = 0; k0 < Tt; k0 += 4) {
        const int k = k0 + 2 * half;
        v2f a, bv;
        a.x  = aptr[(size_t)k * aK];
        a.y  = aptr[(size_t)(k + 1) * aK];
        bv.x = bptr[(size_t)k * bK];
        bv.y = bptr[(size_t)(k + 1) * bK];
        acc = __builtin_amdgcn_wmma_f32_16x16x4_f32(
            false, a, false, bv, (short)0, acc, false, false);
    }

    float* out = TR + (size_t)b * OUTF * INF
                    + (size_t)(tm * 16 + 8 * half) * INF + tn * 16 + row;
#pragma unroll
    for (int r = 0; r < 8; ++r)
        out[(size_t)r * INF] = acc[r];
}

// ---------------------------------------------------------------------------
extern "C" void kernel_launch(void* const* d_in, const int* in_sizes, int n_in,
                              void* d_out, int out_size, void* d_ws, size_t ws_size,
                              hipStream_t stream) {
    const float* X = (const float*)d_in[0];    // (T, B, IN) spikes
    const float* W = (const float*)d_in[1];    // (OUT, IN)

    float* out    = (float*)d_out;
    float* S_out  = out;                                   // (T, B, OUT)
    float* U_hist = out + (size_t)Tt * BO;                 // (2T, B, OUT)
    float* TR     = U_hist + (size_t)2 * Tt * BO;          // (B, OUT, IN)

    float* C = (float*)d_ws;                               // (T, B, OUT) = 26.2 MB

    // Phase 1: 800 x 32 tiles, 4 waves/block -> exact grid, EXEC all-ones.
    lif_gemm_xw<<<(Tt * Bb / 16) * (OUTF / 16) / 4, 128, 0, stream>>>(X, W, C);
    // Phase 2: 65536 threads.
    lif_scan<<<BO / 256, 256, 0, stream>>>(C, S_out, U_hist);
    // Phase 3: 128 * 1024 tiles, 8 waves/block -> exact grid.
    lif_trace<<<(Bb * 32 * 32) / 8, 256, 0, stream>>>(C, X, TR);
}